// IAG_26749056319907
// MI455X (gfx1250) — compile-verified
//
#include <hip/hip_runtime.h>

typedef __attribute__((ext_vector_type(2))) float v2f;
typedef __attribute__((ext_vector_type(8))) float v8f;

// D = A(16x4) * B(4x16) + C, fp32, wave32.
#define WMMA_F32(a, b, c) \
  __builtin_amdgcn_wmma_f32_16x16x4_f32(false, (a), false, (b), (short)0, (c), false, false)

// Async global->LDS copy (4B per lane), tracked by ASYNCcnt.
// LDS byte offset = low 32 bits of the generic pointer (ISA aperture rule).
__device__ __forceinline__ void async_g2l_b32(const void* g, void* lds) {
  unsigned off = (unsigned)(unsigned long long)lds;
  asm volatile("global_load_async_to_lds_b32 %0, %1, off"
               :: "v"(off), "v"(g) : "memory");
}
__device__ __forceinline__ void async_wait0() {
  asm volatile("s_wait_asynccnt 0" ::: "memory");
}

__device__ __forceinline__ v2f lds_v2f(const float* p) {
  return *(const v2f*)p;   // 8B-aligned by construction (even strides/offsets)
}

// ---------------------------------------------------------------------------
// Kernel 1: per-batch  OQ[b] = (P @ x[b] + Bb) @ Q        (128x64 fp32)
// B operands stored K-contiguous (transposed) in LDS -> single ds_load_b64/frag
// ---------------------------------------------------------------------------
__global__ __launch_bounds__(256) void k1_oq(const float* __restrict__ x,
                                             const float* __restrict__ P,
                                             const float* __restrict__ Bb,
                                             const float* __restrict__ Q,
                                             float* __restrict__ OQ) {
  __shared__ __align__(16) float sP [128][130];  // A: row-major (n, k)
  __shared__ __align__(16) float sxT[64][130];   // B^T: (col, k), k = 0..127
  __shared__ __align__(16) float sO [128][66];   // A for phase 2: (n, k)
  __shared__ __align__(16) float sQT[64][66];    // B^T: (col, k), k = 0..63
  __shared__ float sB[128];
  const int b = blockIdx.x;
  const int t = threadIdx.x;

  for (int it = 0; it < 64; ++it) { int e = t + 256 * it; async_g2l_b32(P + e, &sP[e >> 7][e & 127]); }
  for (int it = 0; it < 32; ++it) {
    int e = t + 256 * it; const int k = e >> 6, col = e & 63;
    async_g2l_b32(x + b * 8192 + e, &sxT[col][k]);
  }
  for (int it = 0; it < 16; ++it) {
    int e = t + 256 * it; const int k = e >> 6, col = e & 63;
    async_g2l_b32(Q + e, &sQT[col][k]);
  }
  if (t < 128) sB[t] = Bb[t];
  async_wait0();
  __syncthreads();

  const int w = t >> 5, lane = t & 31, half = lane >> 4, lr = lane & 15;
  const int i = w;  // one 16-row tile band per wave

  // phase 1: O = P @ x[b] + Bb   (K = 128 -> 32 wmma steps)
  {
    v8f c[4] = {};
    for (int kk = 0; kk < 32; ++kk) {
      const int kb = kk * 4 + 2 * half;
      const v2f a = lds_v2f(&sP[i * 16 + lr][kb]);
      for (int j = 0; j < 4; ++j) {
        const v2f bb = lds_v2f(&sxT[j * 16 + lr][kb]);
        c[j] = WMMA_F32(a, bb, c[j]);
      }
    }
    for (int j = 0; j < 4; ++j)
      for (int r = 0; r < 8; ++r) {
        const int row = i * 16 + r + 8 * half, col = j * 16 + lr;
        sO[row][col] = c[j][r] + sB[row];
      }
  }
  __syncthreads();

  // phase 2: OQ = O @ Q   (K = 64 -> 16 wmma steps)
  {
    v8f c[4] = {};
    for (int kk = 0; kk < 16; ++kk) {
      const int kb = kk * 4 + 2 * half;
      const v2f a = lds_v2f(&sO[i * 16 + lr][kb]);
      for (int j = 0; j < 4; ++j) {
        const v2f bb = lds_v2f(&sQT[j * 16 + lr][kb]);
        c[j] = WMMA_F32(a, bb, c[j]);
      }
    }
    for (int j = 0; j < 4; ++j)
      for (int r = 0; r < 8; ++r) {
        const int row = i * 16 + r + 8 * half, col = j * 16 + lr;
        OQ[b * 8192 + row * 64 + col] = c[j][r];
      }
  }
}

// ---------------------------------------------------------------------------
// Kernel 2: per (f, b): A_f = relu(OQ[b] @ T_f) in LDS, degree-normalize,
// Horner-apply (I + A + A^2 + A^3) to x[b,:,f], emit Xf[b,:,f].
// ---------------------------------------------------------------------------
__global__ __launch_bounds__(256) void k2_prop(const float* __restrict__ x,
                                               const float* __restrict__ T,
                                               const float* __restrict__ OQ,
                                               float* __restrict__ Xf) {
  __shared__ __align__(16) float sTT[128][66];   // B^T: (m, k) = T[k, m*64+f]
  __shared__ __align__(16) float sOQ[128][66];   // A: (n, k)
  __shared__ __align__(16) float sA [128][129];  // odd pad: conflict-free walks
  __shared__ float rs[128], cs[128];
  __shared__ float sxf[128], sv[128];
  const int f = blockIdx.x, b = blockIdx.y;
  const int t = threadIdx.x;

  for (int it = 0; it < 32; ++it) { int e = t + 256 * it; async_g2l_b32(OQ + b * 8192 + e, &sOQ[e >> 6][e & 63]); }
  for (int it = 0; it < 32; ++it) {
    int e = t + 256 * it;
    const int k = e >> 7, m = e & 127;
    async_g2l_b32(T + k * 8192 + m * 64 + f, &sTT[m][k]);
  }
  if (t < 128) { const float xv = x[b * 8192 + t * 64 + f]; sxf[t] = xv; sv[t] = xv; }
  async_wait0();
  __syncthreads();

  const int w = t >> 5, lane = t & 31, half = lane >> 4, lr = lane & 15;
  const int i = w;  // 8x8 tiles of 16x16; wave w owns tile-row w

  {
    v8f c[8] = {};
    for (int kk = 0; kk < 16; ++kk) {   // K = 64
      const int kb = kk * 4 + 2 * half;
      const v2f a = lds_v2f(&sOQ[i * 16 + lr][kb]);
      for (int j = 0; j < 8; ++j) {
        const v2f bb = lds_v2f(&sTT[j * 16 + lr][kb]);
        c[j] = WMMA_F32(a, bb, c[j]);
      }
    }
    for (int j = 0; j < 8; ++j)
      for (int r = 0; r < 8; ++r) {     // fused ReLU on store
        const int row = i * 16 + r + 8 * half, col = j * 16 + lr;
        sA[row][col] = fmaxf(c[j][r], 0.0f);
      }
  }
  __syncthreads();

  // row sums (threads 0..127) and column sums (threads 128..255); store rsqrt
  if (t < 128) {
    float s = 0.0f;
    for (int m = 0; m < 128; ++m) s += sA[t][m];
    rs[t] = rsqrtf(s);
  } else {
    const int m = t - 128;
    float s = 0.0f;
    for (int n = 0; n < 128; ++n) s += sA[n][m];
    cs[m] = rsqrtf(s);
  }
  __syncthreads();

  // A[n][m] *= rsqrt(colsum[n]) * rsqrt(rowsum[m])   (reference's indexing)
  for (int it = 0; it < 64; ++it) {
    const int e = t * 64 + it;
    const int n = e >> 7, m = e & 127;
    sA[n][m] *= cs[n] * rs[m];
  }
  __syncthreads();

  // Horner: v <- xf + A v, 3 times  ==> v = (I + A + A^2 + A^3) xf
  for (int it = 0; it < 3; ++it) {
    float acc = 0.0f;
    if (t < 128) {
      for (int m = 0; m < 128; ++m) acc += sA[t][m] * sv[m];
    }
    __syncthreads();
    if (t < 128) sv[t] = sxf[t] + acc;
    __syncthreads();
  }
  if (t < 128) Xf[b * 8192 + t * 64 + f] = sv[t];
}

// ---------------------------------------------------------------------------
// Kernel 3: out[b] = Xf[b] @ U      (128x64 @ 64x64)
// ---------------------------------------------------------------------------
__global__ __launch_bounds__(256) void k3_out(const float* __restrict__ Xf,
                                              const float* __restrict__ U,
                                              float* __restrict__ out) {
  __shared__ __align__(16) float sX [128][66];   // A: (n, k)
  __shared__ __align__(16) float sUT[64][66];    // B^T: (col, k)
  const int b = blockIdx.x, t = threadIdx.x;

  for (int it = 0; it < 32; ++it) { int e = t + 256 * it; async_g2l_b32(Xf + b * 8192 + e, &sX[e >> 6][e & 63]); }
  for (int it = 0; it < 16; ++it) {
    int e = t + 256 * it; const int k = e >> 6, col = e & 63;
    async_g2l_b32(U + e, &sUT[col][k]);
  }
  async_wait0();
  __syncthreads();

  const int w = t >> 5, lane = t & 31, half = lane >> 4, lr = lane & 15;
  const int i = w;

  v8f c[4] = {};
  for (int kk = 0; kk < 16; ++kk) {   // K = 64
    const int kb = kk * 4 + 2 * half;
    const v2f a = lds_v2f(&sX[i * 16 + lr][kb]);
    for (int j = 0; j < 4; ++j) {
      const v2f bb = lds_v2f(&sUT[j * 16 + lr][kb]);
      c[j] = WMMA_F32(a, bb, c[j]);
    }
  }
  for (int j = 0; j < 4; ++j)
    for (int r = 0; r < 8; ++r) {
      const int row = i * 16 + r + 8 * half, col = j * 16 + lr;
      out[b * 8192 + row * 64 + col] = c[j][r];
    }
}

// ---------------------------------------------------------------------------
extern "C" void kernel_launch(void* const* d_in, const int* in_sizes, int n_in,
                              void* d_out, int out_size, void* d_ws, size_t ws_size,
                              hipStream_t stream) {
  const float* x  = (const float*)d_in[0];  // (32,128,64)
  const float* P  = (const float*)d_in[1];  // (128,128)
  const float* Bb = (const float*)d_in[2];  // (128,1)
  const float* Q  = (const float*)d_in[3];  // (64,64)
  const float* T  = (const float*)d_in[4];  // (64,8192)
  const float* U  = (const float*)d_in[5];  // (64,64)
  float* out = (float*)d_out;               // (32,128,64)

  float* OQ = (float*)d_ws;                 // 32*128*64 floats = 1 MB
  float* Xf = OQ + 32 * 128 * 64;           // 32*128*64 floats = 1 MB

  k1_oq<<<32, 256, 0, stream>>>(x, P, Bb, Q, OQ);
  k2_prop<<<dim3(64, 32), 256, 0, stream>>>(x, T, OQ, Xf);
  k3_out<<<32, 256, 0, stream>>>(Xf, U, out);
}